// joint_network_25589415149982
// MI455X (gfx1250) — compile-verified
//
#include <hip/hip_runtime.h>
#include <hip/hip_bf16.h>
#include <math.h>

// ---------------------------------------------------------------------------
// SNN with Hebbian plasticity, T=32, B=32, N=1, D=1024.
//   base[t,b,:] = x@W1^T + b1 + x2@W2^T + b2          (fp32 WMMA GEMM, once)
//   per step t: state = base + alpha * x.bmm(hebb)    (deterministic 4-way
//               mem/spike/tanh pointwise               partial reduction)
//               hebb  = clip(0.9*hebb + beta*x (x) g, -4, 4)
// hebb update(t) fused with bmm read(t+1): one 128MB R + 128MB W pass/step.
// hebb (128MB) is kept L2-resident on MI455X (192MB L2); hebb stores use
// TH_STORE_WB so dirty lines stay in L2 instead of write-rinsing to HBM.
// ---------------------------------------------------------------------------

#define D      1024
#define B      32
#define T      32
#define THRESH 0.35f
#define DECAY  0.5f
#define WDECAY 0.9f

typedef __attribute__((ext_vector_type(2))) float v2f;
typedef __attribute__((ext_vector_type(4))) float v4f;
typedef __attribute__((ext_vector_type(8))) float v8f;

// -------------------------------------------------------------- zero scratch
__global__ __launch_bounds__(256) void k_zero(float* __restrict__ p, int n) {
  int i = blockIdx.x * 256 + threadIdx.x;
  if (i < n) p[i] = 0.0f;
}

// ------------------------------------------------- 1024x1024 f32 transpose
__global__ __launch_bounds__(256) void k_transpose(const float* __restrict__ in,
                                                   float* __restrict__ out) {
  __shared__ float tile[32][33];
  const int tx = threadIdx.x;             // 0..31
  const int x  = blockIdx.x * 32 + tx;
  const int y0 = blockIdx.y * 32;
  for (int i = threadIdx.y; i < 32; i += 8)
    tile[i][tx] = in[(y0 + i) * D + x];
  __syncthreads();
  const int ox = blockIdx.y * 32 + tx;
  for (int i = threadIdx.y; i < 32; i += 8)
    out[(blockIdx.x * 32 + i) * D + ox] = tile[tx][i];
}

// ---------------------------------------------------------------------------
// base[row,e] = sum_d X[row,d]*W1T[d,e] + sum_d X2[row,d]*W2T[d,e] + b1[e]+b2[e]
// One wave per 16x16 C tile, V_WMMA_F32_16X16X4_F32 over K=1024.
// A 16x4 layout : lane L holds M=L%16 ; VGPR j holds K = j + 2*(L/16)
// B  4x16 layout: lane L holds N=L%16 ; VGPR j holds K = j + 2*(L/16)
// C/D          : VGPR r, lane L -> M = r + 8*(L/16), N = L%16
// ---------------------------------------------------------------------------
__global__ __launch_bounds__(32) void k_gemm(const float* __restrict__ X,
                                             const float* __restrict__ X2,
                                             const float* __restrict__ W1T,
                                             const float* __restrict__ W2T,
                                             const float* __restrict__ b1,
                                             const float* __restrict__ b2,
                                             float* __restrict__ base) {
  const int lane = threadIdx.x;
  const int half = lane >> 4;        // 0 | 1
  const int lr   = lane & 15;
  const int m0   = blockIdx.x * 16;
  const int e0   = blockIdx.y * 16;

  v8f acc = {};

  const float* a1 = X  + (size_t)(m0 + lr) * D + 2 * half;   // + k0 -> K
  const float* a2 = X2 + (size_t)(m0 + lr) * D + 2 * half;
  const float* p1 = W1T + (size_t)(2 * half) * D + e0 + lr;  // + k0*D -> K row
  const float* p2 = W2T + (size_t)(2 * half) * D + e0 + lr;

  for (int k0 = 0; k0 < D; k0 += 4) {
    v2f a, b;
    a[0] = a1[k0];
    a[1] = a1[k0 + 1];
    b[0] = p1[(size_t)k0 * D];
    b[1] = p1[(size_t)k0 * D + D];
    acc = __builtin_amdgcn_wmma_f32_16x16x4_f32(false, a, false, b,
                                                (short)0, acc, false, false);
  }
  for (int k0 = 0; k0 < D; k0 += 4) {
    v2f a, b;
    a[0] = a2[k0];
    a[1] = a2[k0 + 1];
    b[0] = p2[(size_t)k0 * D];
    b[1] = p2[(size_t)k0 * D + D];
    acc = __builtin_amdgcn_wmma_f32_16x16x4_f32(false, a, false, b,
                                                (short)0, acc, false, false);
  }

  const float bias = b1[e0 + lr] + b2[e0 + lr];
#pragma unroll
  for (int r = 0; r < 8; ++r)
    base[(size_t)(m0 + r + 8 * half) * D + e0 + lr] = acc[r] + bias;
}

// ---------------------------------------------------------------------------
// Per-step pointwise: reduce 4 d-chunk partials, membrane update, spike out,
// g = tanh(mem/THRESH - eta). One thread per (b,e).  32K threads.
// ---------------------------------------------------------------------------
__global__ __launch_bounds__(256) void k_point(const float* __restrict__ base,
                                               const float* __restrict__ partials,
                                               const float* __restrict__ mem0,
                                               const float* __restrict__ spike0,
                                               const float* __restrict__ alpha,
                                               const float* __restrict__ eta,
                                               float* __restrict__ mem,
                                               float* __restrict__ g,
                                               float* __restrict__ out_spikes,
                                               int t) {
  const int idx = blockIdx.x * 256 + threadIdx.x;   // 0 .. 32767
  const int b = idx >> 10;
  const int e = idx & (D - 1);

  float st = base[(size_t)(t * B + b) * D + e];
  float p  = partials[(size_t)(b * 4 + 0) * D + e]
           + partials[(size_t)(b * 4 + 1) * D + e]
           + partials[(size_t)(b * 4 + 2) * D + e]
           + partials[(size_t)(b * 4 + 3) * D + e];
  st += alpha[0] * p;

  float m_prev, s_prev;
  if (t == 0) {
    m_prev = mem0[idx];
    s_prev = spike0[idx];
  } else {
    m_prev = mem[idx];
    s_prev = (m_prev > THRESH) ? 1.0f : 0.0f;   // == spike carried from t-1
  }
  const float m = (m_prev - s_prev * THRESH) * DECAY + st;
  mem[idx] = m;
  out_spikes[(size_t)t * (B * D) + idx] = (m > THRESH) ? 1.0f : 0.0f;
  g[idx] = tanhf(m / THRESH - eta[0]);
}

// ---------------------------------------------------------------------------
// Fused: hebb(t) = clip(0.9*hebb + (beta*x_t) (x) g_t), plus partials for
// step t+1:  partial[b,dc,e] = sum_{d in chunk} x_{t+1}[d]*hebb_new[d,e].
// Grid (b, d-chunk, e-chunk); 64 threads x 4 e each (b128 streams); loop 256 d.
// hebb rows are e-contiguous -> 512B/wave coalesced lines, L2 resident.
// Stores use TH_STORE_WB so the 128MB working set stays dirty-in-L2 across
// all 32 sweeps instead of write-rinsing to HBM.
// ---------------------------------------------------------------------------
__global__ __launch_bounds__(64) void k_hebb(const float* __restrict__ hebb_src,
                                             float* __restrict__ hebb_dst,
                                             const float* __restrict__ x_seq,
                                             const float* __restrict__ g,
                                             const float* __restrict__ beta,
                                             float* __restrict__ partials,
                                             int t, int tn) {
  const int b  = blockIdx.x;
  const int dc = blockIdx.y;                    // d chunk, 0..3
  const int ec = blockIdx.z;                    // e chunk, 0..3
  const int e4 = ec * 256 + threadIdx.x * 4;    // 4 consecutive e per thread
  const int d0 = dc * 256;

  __shared__ float bx[256];   // beta * x_t[b, d0..d0+255]
  __shared__ float xn[256];   // x_{t+1}[b, d0..d0+255]
  const float be = beta[0];
  for (int i = threadIdx.x; i < 256; i += 64) {
    bx[i] = be * x_seq[(size_t)(t  * B + b) * D + d0 + i];
    xn[i] =      x_seq[(size_t)(tn * B + b) * D + d0 + i];
  }
  __syncthreads();

  const v4f ge = *(const v4f*)(g + b * D + e4);
  const float* src = hebb_src + (size_t)b * D * D + (size_t)d0 * D + e4;
  float*       dst = hebb_dst + (size_t)b * D * D + (size_t)d0 * D + e4;

  v4f acc = {};
#pragma unroll 4
  for (int d = 0; d < 256; ++d) {
    v4f h = *(const v4f*)(src + (size_t)d * D);
    const float s = bx[d];
    const float xv = xn[d];
#pragma unroll
    for (int j = 0; j < 4; ++j) {
      float hv = WDECAY * h[j] + s * ge[j];
      hv = fminf(4.0f, fmaxf(-4.0f, hv));
      h[j] = hv;
      acc[j] += xv * hv;
    }
    // hebb fits in MI455X's 192MB L2: force dirty-in-far-cache (WB) instead
    // of the default RT write-rinse so replays of this sweep stay in L2.
    const float* p = dst + (size_t)d * D;
    asm volatile("global_store_b128 %0, %1, off th:TH_STORE_WB scope:SCOPE_DEV"
                 :: "v"(p), "v"(h) : "memory");
  }
  *(v4f*)(partials + (size_t)(b * 4 + dc) * D + e4) = acc;
}

// ---------------------------------------------------------------------------
extern "C" void kernel_launch(void* const* d_in, const int* in_sizes, int n_in,
                              void* d_out, int out_size, void* d_ws, size_t ws_size,
                              hipStream_t stream) {
  const float* x      = (const float*)d_in[0];   // [T,B,1,D]
  const float* x2     = (const float*)d_in[1];   // [T,B,1,D]
  const float* W1     = (const float*)d_in[2];   // [D,D]
  const float* b1     = (const float*)d_in[3];   // [D]
  const float* W2     = (const float*)d_in[4];   // [D,D]
  const float* b2     = (const float*)d_in[5];   // [D]
  const float* alpha  = (const float*)d_in[6];   // [1]
  const float* beta   = (const float*)d_in[7];   // [1]
  const float* eta    = (const float*)d_in[8];   // [1]
  const float* mem0   = (const float*)d_in[9];   // [B,1,D]
  const float* spike0 = (const float*)d_in[10];  // [B,1,D]
  const float* hebb0  = (const float*)d_in[11];  // [B,D,D]
  float* out = (float*)d_out;                    // [T,B,1,D]

  float* ws       = (float*)d_ws;
  float* W1T      = ws;                          // 1M floats
  float* W2T      = ws + 1048576;                // 1M
  float* base     = ws + 2097152;                // 1M
  float* mem      = ws + 3145728;                // 32K
  float* g        = ws + 3178496;                // 32K
  float* partials = ws + 3211264;                // 128K (B x 4 x D)
  float* hebb     = ws + 3342336;                // 32M  (B x D x D)

  // partials must be zero for t=0 (hebb0 == 0 -> bmm contribution is zero)
  k_zero<<<512, 256, 0, stream>>>(partials, B * 4 * D);

  dim3 tb(32, 8);
  k_transpose<<<dim3(32, 32), tb, 0, stream>>>(W1, W1T);
  k_transpose<<<dim3(32, 32), tb, 0, stream>>>(W2, W2T);

  // base = x@W1^T + x2@W2^T + b1 + b2  (fp32 WMMA, 64x64 tiles of 16x16)
  k_gemm<<<dim3(64, 64), 32, 0, stream>>>(x, x2, W1T, W2T, b1, b2, base);

  for (int t = 0; t < T; ++t) {
    k_point<<<(B * D) / 256, 256, 0, stream>>>(base, partials, mem0, spike0,
                                               alpha, eta, mem, g, out, t);
    const int tn = (t < T - 1) ? t + 1 : T - 1;          // t=31 result unused
    const float* hsrc = (t == 0) ? hebb0 : hebb;         // hebb0 read in place
    k_hebb<<<dim3(B, 4, 4), 64, 0, stream>>>(hsrc, hebb, x, g, beta,
                                             partials, t, tn);
  }
}